// TransferNet_8246337208853
// MI455X (gfx1250) — compile-verified
//
#include <hip/hip_runtime.h>
#include <stdint.h>

// ---------------- problem constants (fixed by the reference) ----------------
constexpr int Tt  = 1024;   // time steps
constexpr int Cc  = 6;      // channels
// ---------------- tiling ----------------
constexpr int NB     = 64;  // batch rows per block == threads per block (2 wave32)
constexpr int CH     = 16;  // time steps per TDM chunk
constexpr int SEGLEN = 64;  // time steps owned per segment
constexpr int WARM   = 16;  // warm-up steps (contraction ~0.08/step -> 1e-17 after 16)
constexpr int NSEG   = Tt / SEGLEN;
static_assert(WARM % CH == 0, "warmup must be chunk aligned");
static_assert(SEGLEN % CH == 0, "segment must be chunk aligned");

constexpr int ROW_ELEMS = CH * Cc;                    // 96 dwords of data per row
constexpr int ROW_DW    = ROW_ELEMS + ROW_ELEMS / 32; // TDM pad: +1 dword per 32 -> 99
constexpr int BUF_DW    = NB * ROW_DW;                // 6336 dwords per buffer

typedef unsigned int u32;
typedef u32 v4u __attribute__((ext_vector_type(4)));
typedef int v8i __attribute__((ext_vector_type(8)));
typedef int v4i __attribute__((ext_vector_type(4)));

// Issue one 2D TDM tile load: NB rows x (CH*Cc) f32 elements, row stride T*C
// elements, into LDS at lds_byte with 1-dword padding every 32 dwords
// (makes the LDS row stride 99 dwords -> bank-conflict-free lane access).
__device__ __forceinline__ void tdm_load_tile(u32 lds_byte, const float* gp) {
  uint64_t ga = (uint64_t)(uintptr_t)gp;
  v4u g0;
  g0[0] = 1u;                                            // count=1 (valid), user mode
  g0[1] = (u32)__builtin_amdgcn_readfirstlane((int)lds_byte);           // lds_addr
  g0[2] = (u32)__builtin_amdgcn_readfirstlane((int)(u32)ga);            // gaddr[31:0]
  g0[3] = (u32)__builtin_amdgcn_readfirstlane(
      (int)((u32)(ga >> 32) | (2u << 30)));              // gaddr[56:32] | type=2
  v8i g1;
  g1[0] = (2 << 16)        // data_size = 4 bytes
        | (1 << 20)        // pad_enable
        | (4 << 22)        // pad_interval: every 32 dwords
        | (0 << 25);       // pad_amount: 1 dword
  g1[1] = (ROW_ELEMS & 0xFFFF) << 16;                    // tensor_dim0[15:0]
  g1[2] = (NB & 0xFFFF) << 16;                           // dim0 hi=0 | tensor_dim1 lo
  g1[3] = (ROW_ELEMS & 0xFFFF) << 16;                    // dim1 hi=0 | tile_dim0
  g1[4] = NB;                                            // tile_dim1 | tile_dim2=0
  g1[5] = Tt * Cc;                                       // tensor_dim0_stride lo (6144)
  g1[6] = 0;                                             // stride hi | dim1_stride lo
  g1[7] = 0;
  v4i z4 = {0, 0, 0, 0};                                 // groups 2/3 unused (2D tile)
  v8i z8 = {0, 0, 0, 0, 0, 0, 0, 0};                     // extra operand (clang-23 form)
  __builtin_amdgcn_tensor_load_to_lds(g0, g1, z4, z4, z8, 0);
}

// LDS read honoring the TDM pad layout (folds to immediates when e is constant).
__device__ __forceinline__ float lds_row_at(const float* __restrict__ Lrow, int e) {
  return Lrow[e + (e >> 5)];
}

// One recurrence step: ns_j = a*cur_j + (1-a)*ln2*log2( sum_i 2^(s_i*Tl_ij) )
// where Tl = softmax(trans)*log2(e) and c2 = (1-a)*ln2.
// Arguments are range-bounded (|s*Tl| <~ 2, sum in [1.5, 24]), so the raw
// hardware v_exp_f32 / v_log_f32 builtins are safe (no denorm/overflow guards).
__device__ __forceinline__ void do_step(const float* __restrict__ Lrow, int tl,
                                        const float* __restrict__ Tl, float a,
                                        float c2, float s[6],
                                        float* __restrict__ ochunk, bool dowrite) {
  float cur[6];
#pragma unroll
  for (int c = 0; c < 6; ++c) {
    int e = tl * 6 + c;
    cur[c] = lds_row_at(Lrow, e);
  }
  float ns[6];
#pragma unroll
  for (int j = 0; j < 6; ++j) {
    float sum = 0.0f;
#pragma unroll
    for (int i = 0; i < 6; ++i)
      sum += __builtin_amdgcn_exp2f(s[i] * Tl[i * 6 + j]);   // bare v_exp_f32
    ns[j] = fmaf(a, cur[j], c2 * __builtin_amdgcn_logf(sum)); // bare v_log_f32 (log2)
  }
#pragma unroll
  for (int c = 0; c < 6; ++c) s[c] = ns[c];
  if (dowrite) {
    float* o = ochunk + tl * 6;   // 24B-aligned -> b64 stores
    ((float2*)o)[0] = make_float2(ns[0], ns[1]);
    ((float2*)o)[1] = make_float2(ns[2], ns[3]);
    ((float2*)o)[2] = make_float2(ns[4], ns[5]);
  }
}

__global__ __launch_bounds__(NB) void transfer_scan_kernel(
    const float* __restrict__ feats, const float* __restrict__ alpha,
    const float* __restrict__ trans, float* __restrict__ out) {
  __shared__ float lds[2 * BUF_DW];   // 50688 B: two TDM-staged chunks

  const int tid  = threadIdx.x;
  const int b0   = blockIdx.x * NB;
  const int b    = b0 + tid;
  const int seg  = blockIdx.y;
  const int wstart = seg * SEGLEN;                 // first time step we write
  const int t0     = (seg == 0) ? 0 : (wstart - WARM);
  const int tend   = wstart + SEGLEN;
  const int nchunks = (tend - t0) / CH;            // 4 (seg 0) or 5

  // ---- per-lane constants: a = sigmoid(alpha); Tl = softmax(trans)*log2e ----
  // One-time setup uses the accurate libm expf.
  const float aa = alpha[0];
  const float a  = 1.0f / (1.0f + expf(-aa));
  const float c2 = (1.0f - a) * 0.6931471805599453f;   // (1-a)*ln2
  float Tl[36];
  {
    const float log2e = 1.4426950408889634f;
#pragma unroll
    for (int i = 0; i < 6; ++i) {
      float r[6];
      float m = -3.4e38f;
#pragma unroll
      for (int j = 0; j < 6; ++j) { r[j] = trans[i * 6 + j]; m = fmaxf(m, r[j]); }
      float ssum = 0.0f;
#pragma unroll
      for (int j = 0; j < 6; ++j) { r[j] = expf(r[j] - m); ssum += r[j]; }
      const float inv = 1.0f / ssum;
#pragma unroll
      for (int j = 0; j < 6; ++j) Tl[i * 6 + j] = r[j] * inv * log2e;
    }
  }

  const u32 lds_base_byte = (u32)(uintptr_t)(void*)&lds[0];

  // ---- prologue: DMA chunk 0 ----
  if (tid == 0) {
    tdm_load_tile(lds_base_byte, feats + ((long)b0 * Tt + t0) * Cc);
    __builtin_amdgcn_s_wait_tensorcnt(0);
  }
  __syncthreads();

  float s[6];
  for (int k = 0; k < nchunks; ++k) {
    const int tbase = t0 + k * CH;
    // async prefetch of next chunk into the other buffer
    if (tid == 0 && (k + 1) < nchunks)
      tdm_load_tile(lds_base_byte + (u32)(((k + 1) & 1) * BUF_DW * 4),
                    feats + ((long)b0 * Tt + tbase + CH) * Cc);

    const float* Lrow  = lds + (k & 1) * BUF_DW + tid * ROW_DW;
    float* ochunk      = out + ((long)b * Tt + tbase) * Cc;
    const bool dowrite = (tbase >= wstart);

    if (k == 0) {
      // seed state from feats[:, t0] (exact for seg 0; warm-up seed otherwise)
#pragma unroll
      for (int c = 0; c < 6; ++c) s[c] = lds_row_at(Lrow, c);
      if (dowrite) {  // seg 0 only: out[:,0] = feats[:,0]
        ((float2*)ochunk)[0] = make_float2(s[0], s[1]);
        ((float2*)ochunk)[1] = make_float2(s[2], s[3]);
        ((float2*)ochunk)[2] = make_float2(s[4], s[5]);
      }
#pragma unroll
      for (int tl = 1; tl < CH; ++tl) do_step(Lrow, tl, Tl, a, c2, s, ochunk, dowrite);
    } else {
#pragma unroll
      for (int tl = 0; tl < CH; ++tl) do_step(Lrow, tl, Tl, a, c2, s, ochunk, dowrite);
    }

    if (tid == 0 && (k + 1) < nchunks) __builtin_amdgcn_s_wait_tensorcnt(0);
    __syncthreads();
  }
}

extern "C" void kernel_launch(void* const* d_in, const int* in_sizes, int n_in,
                              void* d_out, int out_size, void* d_ws, size_t ws_size,
                              hipStream_t stream) {
  (void)n_in; (void)out_size; (void)d_ws; (void)ws_size;
  const float* feats = (const float*)d_in[0];
  const float* alpha = (const float*)d_in[1];
  const float* trans = (const float*)d_in[2];
  float* out = (float*)d_out;

  const int B = in_sizes[0] / (Tt * Cc);   // 8192
  dim3 grid(B / NB, NSEG);                 // 128 x 16 = 2048 blocks
  dim3 block(NB);                          // 64 threads = 2 wave32
  transfer_scan_kernel<<<grid, block, 0, stream>>>(feats, alpha, trans, out);
}